// TemporalActivityRegularizer_7885559955975
// MI455X (gfx1250) — compile-verified
//
#include <hip/hip_runtime.h>
#include <stddef.h>
#include <stdint.h>

// ---------------------------------------------------------------------------
// TemporalActivityRegularizer for MI455X (gfx1250, wave32).
//
// Memory-bound: ~442 MB traffic/call (205MB history read + 205MB history
// write dominate) vs ~2 MFLOP -> pure HBM bandwidth problem (~19us @23.3TB/s).
// Bulk history copy uses the CDNA5 async global<->LDS DMA path (ASYNCcnt):
// data never round-trips through VGPRs, 16 transfers in flight per lane per
// direction, and the per-op global/LDS advance is folded into the async
// instruction's INST_OFFSET (ISA: offset is added to BOTH the LDS and the
// global address), so the steady-state loop is just 16 back-to-back async
// loads, one s_wait_asynccnt, 16 async stores.
//
// d_out layout (reference tuple, flat):
//   [activations 4096*512][loss 1][new_history 100001*512]
// -> new_history base is at an ODD float offset (4B-aligned only), so the
//    copy and the scatter use B32 granularity (a wave32 of b32 ops still
//    moves a full 128B line per instruction -> same coalescing/bandwidth).
// ---------------------------------------------------------------------------

#define B_ROWS     4096
#define DIM        512
#define MAX_ITEMS  100000
#define ONE_MINUS_MOMENT 0.1f

// ---------------- bulk copy: history -> out_history via async LDS DMA ------
#define CP_THREADS 256
#define CP_DEPTH   16                       // async ops in flight per lane
#define CP_CHUNK   (CP_THREADS * CP_DEPTH)  // 4096 floats per block
#define CP_STRIDE  (CP_THREADS * 4)         // 1024 B per depth step

__global__ __launch_bounds__(CP_THREADS)
void hist_copy_async(const float* __restrict__ src, float* __restrict__ dst, int n) {
    __shared__ float buf[CP_CHUNK];          // 16 KB staging, lane-private slots
    const int tid  = threadIdx.x;
    const int base = (int)blockIdx.x * CP_CHUNK;

    const unsigned lds = (unsigned)(size_t)&buf[tid];  // LDS byte addr of slot 0
    const unsigned off = (unsigned)(base + tid) * 4u;  // global byte offset

    if (base + CP_CHUNK <= n) {
        // ---- fast path (12500 of 12501 blocks): no guards, no per-op VALU.
        // INST_OFFSET advances both the LDS slot and the global address.
#pragma unroll
        for (int d = 0; d < CP_DEPTH; ++d) {
            asm volatile("global_load_async_to_lds_b32 %0, %1, %2 offset:%c3"
                         :: "v"(lds), "v"(off), "s"(src), "n"(d * CP_STRIDE)
                         : "memory");
        }
        asm volatile("s_wait_asynccnt 0" ::: "memory");
#pragma unroll
        for (int d = 0; d < CP_DEPTH; ++d) {
            asm volatile("global_store_async_from_lds_b32 %0, %1, %2 offset:%c3"
                         :: "v"(off), "v"(lds), "s"(dst), "n"(d * CP_STRIDE)
                         : "memory");
        }
    } else {
        // ---- guarded tail path (last block only).
#pragma unroll
        for (int d = 0; d < CP_DEPTH; ++d) {
            if (base + d * CP_THREADS + tid < n) {
                asm volatile("global_load_async_to_lds_b32 %0, %1, %2 offset:%c3"
                             :: "v"(lds), "v"(off), "s"(src), "n"(d * CP_STRIDE)
                             : "memory");
            }
        }
        asm volatile("s_wait_asynccnt 0" ::: "memory");
#pragma unroll
        for (int d = 0; d < CP_DEPTH; ++d) {
            if (base + d * CP_THREADS + tid < n) {
                asm volatile("global_store_async_from_lds_b32 %0, %1, %2 offset:%c3"
                             :: "v"(off), "v"(lds), "s"(dst), "n"(d * CP_STRIDE)
                             : "memory");
            }
        }
    }
    // Drain async stores before wave end (S_ENDPGM's implicit wait-idle would
    // also cover this; explicit is belt-and-braces).
    asm volatile("s_wait_asynccnt 0" ::: "memory");
}

// ---------------- fused gather + mse + scatter + activation passthrough ----
__device__ __forceinline__ void atomic_add_f32(float* p, float v) {
    // relaxed/agent scope -> native global_atomic_add_f32 (no CAS loop)
    __hip_atomic_fetch_add(p, v, __ATOMIC_RELAXED, __HIP_MEMORY_SCOPE_AGENT);
}

__global__ __launch_bounds__(256)
void gather_mse_scatter(const float* __restrict__ act,
                        const float* __restrict__ hist,
                        const int*   __restrict__ samples,
                        float* __restrict__ out_act,
                        float* __restrict__ out_hist,   // 4B-aligned only!
                        float* __restrict__ partial) {
    __shared__ float red[256];
    const int tid  = threadIdx.x;
    const int lane = tid & 31;
    const int wave = tid >> 5;
    const int row  = (int)blockIdx.x * 8 + wave;   // one wave32 per sample row

    const int s   = samples[row];
    const bool on = (s > 0) && (s < MAX_ITEMS);    // strict, matches reference
    const int id  = (s < MAX_ITEMS) ? s : MAX_ITEMS;

    const float4* arow  = (const float4*)(act  + (size_t)row * DIM);
    const float4* hrow  = (const float4*)(hist + (size_t)id  * DIM);
    float4*       oarow = (float4*)(out_act + (size_t)row * DIM);
    float*        ohrow = out_hist + (size_t)id * DIM;

    float acc = 0.0f;
#pragma unroll
    for (int k = 0; k < DIM / (32 * 4); ++k) {     // 4 float4 per lane
        const int c4 = lane + k * 32;              // coalesced across the wave
        const float4 a = arow[c4];
        const float4 h = hrow[c4];
        oarow[c4] = a;                              // activations passthrough
        const float dx = h.x - a.x, dy = h.y - a.y;
        const float dz = h.z - a.z, dw = h.w - a.w;
        acc += dx * dx + dy * dy + dz * dz + dw * dw;
        if (on) {                                   // wave-uniform branch
            const int c = c4 * 4;
            atomic_add_f32(ohrow + c + 0, -ONE_MINUS_MOMENT * dx);
            atomic_add_f32(ohrow + c + 1, -ONE_MINUS_MOMENT * dy);
            atomic_add_f32(ohrow + c + 2, -ONE_MINUS_MOMENT * dz);
            atomic_add_f32(ohrow + c + 3, -ONE_MINUS_MOMENT * dw);
        }
    }

    // Deterministic fixed-tree block reduction of masked row sums.
    red[tid] = on ? acc : 0.0f;
    __syncthreads();
    for (int st = 128; st > 0; st >>= 1) {
        if (tid < st) red[tid] += red[tid + st];
        __syncthreads();
    }
    if (tid == 0) partial[blockIdx.x] = red[0];
}

// ---------------- deterministic loss finalization --------------------------
__global__ __launch_bounds__(256)
void finalize_loss(const float* __restrict__ partial,
                   const int*   __restrict__ iters,
                   float* __restrict__ loss, int nparts) {
    __shared__ float red[256];
    const int tid = threadIdx.x;
    float v = 0.0f;
    for (int i = tid; i < nparts; i += 256) v += partial[i];
    red[tid] = v;
    __syncthreads();
    for (int st = 128; st > 0; st >>= 1) {
        if (tid < st) red[tid] += red[tid + st];
        __syncthreads();
    }
    if (tid == 0) {
        const float it = (float)iters[0];
        const float wr = it * (1.0f / 1000.0f);
        const float cr = it * (1.0f / 100000.0f);
        const float weight = 0.1f * wr / (1.0f + wr) / (1.0f + cr);
        loss[0] = red[0] * weight * (1.0f / ((float)B_ROWS * (float)DIM));
    }
}

// ---------------------------------------------------------------------------
extern "C" void kernel_launch(void* const* d_in, const int* in_sizes, int n_in,
                              void* d_out, int out_size, void* d_ws, size_t ws_size,
                              hipStream_t stream) {
    const float* act     = (const float*)d_in[0];   // (4096, 512) f32
    const float* hist    = (const float*)d_in[1];   // (100001, 512) f32
    const int*   samples = (const int*)d_in[2];     // (4096,) int
    const int*   iters   = (const int*)d_in[3];     // scalar int

    const int bcnt = in_sizes[2];                   // 4096
    const int dcnt = in_sizes[0] / bcnt;            // 512
    const int hsz  = in_sizes[1];                   // 51,200,512 floats

    float* out      = (float*)d_out;
    float* out_act  = out;                           // [0 .. B*D)
    float* loss     = out + (size_t)bcnt * dcnt;     // [B*D]
    float* out_hist = loss + 1;                      // [B*D+1 ..) -- odd offset
    float* partial  = (float*)d_ws;                  // per-block loss partials

    // 1) Bulk copy history -> out_history via async LDS DMA (ASYNCcnt path).
    const int cpBlocks = (hsz + CP_CHUNK - 1) / CP_CHUNK;   // 12501
    hist_copy_async<<<cpBlocks, CP_THREADS, 0, stream>>>(hist, out_hist, hsz);

    // 2) Fused gather + squared-diff + EMA scatter-add + activation copy.
    const int ublocks = bcnt / 8;                    // 512 (8 rows per block)
    gather_mse_scatter<<<ublocks, 256, 0, stream>>>(act, hist, samples,
                                                    out_act, out_hist, partial);

    // 3) Deterministic loss reduction + weight(iterations).
    finalize_loss<<<1, 256, 0, stream>>>(partial, iters, loss, ublocks);
}